// GNN_14920716387107
// MI455X (gfx1250) — compile-verified
//
#include <hip/hip_runtime.h>

typedef float v2f __attribute__((ext_vector_type(2)));
typedef float v8f __attribute__((ext_vector_type(8)));

#define HIDF 64

// ---------------- utility: zero a float buffer ----------------
__global__ void zero_f32(float* __restrict__ p, long n) {
    long i = (long)blockIdx.x * blockDim.x + threadIdx.x;
    if (i < n) p[i] = 0.0f;
}

// ---------------- degree: deg[col[e]] += ew[e] ----------------
__global__ void deg_kernel(const int* __restrict__ col, const float* __restrict__ ew,
                           float* __restrict__ deg, int E) {
    int e = blockIdx.x * blockDim.x + threadIdx.x;
    if (e < E) atomicAdd(&deg[col[e]], ew[e]);
}

// ---------------- deg -> deg^(-1/2) in place ----------------
__global__ void dinv_kernel(float* __restrict__ deg, int n) {
    int i = blockIdx.x * blockDim.x + threadIdx.x;
    if (i < n) {
        float d = deg[i];
        deg[i] = (d > 0.0f) ? rsqrtf(d) : 0.0f;
    }
}

// ---------------- per-edge norm = dinv[row]*ew*dinv[col] ----------------
__global__ void norm_kernel(const int* __restrict__ row, const int* __restrict__ col,
                            const float* __restrict__ ew, const float* __restrict__ dinv,
                            float* __restrict__ nrm, int E) {
    int e = blockIdx.x * blockDim.x + threadIdx.x;
    if (e < E) nrm[e] = dinv[row[e]] * ew[e] * dinv[col[e]];
}

// ---------------- propagation hop: hout[col] += hin[row]*norm ----------------
// one thread per (edge, feature); lanes within a wave cover consecutive features
// of the same edge -> coalesced gather and coalesced atomic scatter.
template <int F>
__global__ void prop_kernel(const float* __restrict__ hin, const int* __restrict__ row,
                            const int* __restrict__ col, const float* __restrict__ nrm,
                            float* __restrict__ hout, int E) {
    long t = (long)blockIdx.x * blockDim.x + threadIdx.x;
    int e = (int)(t / F);
    int f = (int)(t % F);
    if (e >= E) return;
    float v = hin[(long)row[e] * F + f] * nrm[e];
    atomicAdd(&hout[(long)col[e] * F + f], v);
}

// ---------------- fused 3-term WMMA GEMM + bias + leaky ReLU ----------------
// out[N x 64] = leaky( A0@W[0] + A1@W[1] + A2@W[2] + bias ),  A* are N x F.
// One wave computes one 16x16 C tile with V_WMMA_F32_16X16X4_F32,
// accumulating over all 3 terms and all K-chunks in registers.
template <int F>
__global__ void gemm3_wmma(const float* __restrict__ A0, const float* __restrict__ A1,
                           const float* __restrict__ A2, const float* __restrict__ W,
                           const float* __restrict__ bias, float* __restrict__ out,
                           int nRowTiles) {
    const int lane = threadIdx.x & 31;
    const int wave = threadIdx.x >> 5;
    const int rowTile = blockIdx.x * 2 + (wave >> 2); // 8 waves: 2 row tiles x 4 col tiles
    const int colTile = wave & 3;
    if (rowTile >= nRowTiles) return;

    const int rowBase = rowTile * 16;
    const int colBase = colTile * 16;
    const int half = lane >> 4;   // 0: lanes 0-15, 1: lanes 16-31
    const int l16  = lane & 15;

    v8f c = {};
    const float* As[3] = {A0, A1, A2};
#pragma unroll
    for (int m = 0; m < 3; ++m) {
        const float* __restrict__ A  = As[m];
        const float* __restrict__ Wm = W + (long)m * F * HIDF;
        const float* __restrict__ ar = A + (long)(rowBase + l16) * F + 2 * half;
#pragma unroll
        for (int kc = 0; kc < F / 4; ++kc) {
            const int k0 = kc * 4;
            // A fragment: 16x4 f32 (lane<16: K=0,1 ; lane>=16: K=2,3)
            v2f a;
            a.x = ar[k0];
            a.y = ar[k0 + 1];
            // B fragment: 4x16 f32 (VGPR g: K = g + 2*half, N = l16)
            v2f b;
            b.x = Wm[(long)(k0 + 2 * half) * HIDF + colBase + l16];
            b.y = Wm[(long)(k0 + 2 * half + 1) * HIDF + colBase + l16];
            c = __builtin_amdgcn_wmma_f32_16x16x4_f32(false, a, false, b,
                                                      (short)0, c, false, false);
        }
    }

    const float bv = bias[colBase + l16];
#pragma unroll
    for (int g = 0; g < 8; ++g) {
        float v = c[g] + bv;
        v = (v > 0.0f) ? v : 0.01f * v;
        out[(long)(rowBase + g + 8 * half) * HIDF + colBase + l16] = v;
    }
}

// ---------------- heads: 1000 x (64 -> 3) x 2 ----------------
__global__ void head_kernel(const float* __restrict__ ap, const float* __restrict__ wch,
                            const float* __restrict__ bch, const float* __restrict__ wpw,
                            const float* __restrict__ bpw, float* __restrict__ out,
                            int nAp) {
    int i = blockIdx.x * blockDim.x + threadIdx.x;
    if (i >= nAp) return;
    const float* __restrict__ h = ap + (long)i * HIDF;
    float c0 = bch[0], c1 = bch[1], c2 = bch[2];
    float p0 = bpw[0], p1 = bpw[1], p2 = bpw[2];
#pragma unroll 8
    for (int k = 0; k < HIDF; ++k) {
        float v = h[k];
        c0 += v * wch[k * 3 + 0];
        c1 += v * wch[k * 3 + 1];
        c2 += v * wch[k * 3 + 2];
        p0 += v * wpw[k * 3 + 0];
        p1 += v * wpw[k * 3 + 1];
        p2 += v * wpw[k * 3 + 2];
    }
    out[i * 3 + 0] = c0;
    out[i * 3 + 1] = c1;
    out[i * 3 + 2] = c2;
    out[3 * nAp + i * 3 + 0] = p0;
    out[3 * nAp + i * 3 + 1] = p1;
    out[3 * nAp + i * 3 + 2] = p2;
}

extern "C" void kernel_launch(void* const* d_in, const int* in_sizes, int n_in,
                              void* d_out, int out_size, void* d_ws, size_t ws_size,
                              hipStream_t stream) {
    (void)n_in; (void)out_size; (void)ws_size;

    const float* x   = (const float*)d_in[0];   // (N, 32)
    const int*   ei  = (const int*)  d_in[1];   // (2, E)
    const float* ea  = (const float*)d_in[2];   // (E, 1)
    const float* w1  = (const float*)d_in[3];   // (3, 32, 64)
    const float* b1  = (const float*)d_in[4];   // (64,)
    const float* w2  = (const float*)d_in[5];   // (3, 64, 64)
    const float* b2  = (const float*)d_in[6];   // (64,)
    const float* wch = (const float*)d_in[7];   // (64, 3)
    const float* bch = (const float*)d_in[8];   // (3,)
    const float* wpw = (const float*)d_in[9];   // (64, 3)
    const float* bpw = (const float*)d_in[10];  // (3,)

    const int N = in_sizes[0] / 32;  // 100000
    const int E = in_sizes[2];       // 1600000
    const int* row = ei;
    const int* col = ei + E;

    // scratch layout (floats)
    float* ws   = (float*)d_ws;
    float* deg  = ws;  ws += N;              // N
    float* nrm  = ws;  ws += E;              // E
    float* t1   = ws;  ws += (long)N * 32;   // N x 32 (hop1, layer1)
    float* t2   = ws;  ws += (long)N * 32;   // N x 32 (hop2, layer1)
    float* out1 = ws;  ws += (long)N * 64;   // N x 64 (layer1 output)
    float* t3   = ws;  ws += (long)N * 64;   // N x 64 (hop1, layer2)
    float* t4   = ws;  ws += (long)N * 64;   // N x 64 (hop2, layer2)
    float* out2 = t1;                        // reuse t1+t2 region (N x 64)

    const int B = 256;
    auto blocks = [](long n, int b) { return (int)((n + b - 1) / b); };

    // ---- GCN norm ----
    zero_f32<<<blocks(N, B), B, 0, stream>>>(deg, N);
    deg_kernel<<<blocks(E, B), B, 0, stream>>>(col, ea, deg, E);
    dinv_kernel<<<blocks(N, B), B, 0, stream>>>(deg, N);
    norm_kernel<<<blocks(E, B), B, 0, stream>>>(row, col, ea, deg, nrm, E);

    // ---- Layer 1 (F=32) ----
    zero_f32<<<blocks((long)N * 32, B), B, 0, stream>>>(t1, (long)N * 32);
    prop_kernel<32><<<blocks((long)E * 32, B), B, 0, stream>>>(x, row, col, nrm, t1, E);
    zero_f32<<<blocks((long)N * 32, B), B, 0, stream>>>(t2, (long)N * 32);
    prop_kernel<32><<<blocks((long)E * 32, B), B, 0, stream>>>(t1, row, col, nrm, t2, E);
    {
        int nRowTiles = (N + 15) / 16;                 // 6250
        int nBlocks   = (nRowTiles + 1) / 2;           // 2 row tiles per block (8 waves)
        gemm3_wmma<32><<<nBlocks, 256, 0, stream>>>(x, t1, t2, w1, b1, out1, nRowTiles);
    }

    // ---- Layer 2 (F=64) ----
    zero_f32<<<blocks((long)N * 64, B), B, 0, stream>>>(t3, (long)N * 64);
    prop_kernel<64><<<blocks((long)E * 64, B), B, 0, stream>>>(out1, row, col, nrm, t3, E);
    zero_f32<<<blocks((long)N * 64, B), B, 0, stream>>>(t4, (long)N * 64);
    prop_kernel<64><<<blocks((long)E * 64, B), B, 0, stream>>>(t3, row, col, nrm, t4, E);
    {
        int nRowTiles = (N + 15) / 16;
        int nBlocks   = (nRowTiles + 1) / 2;
        gemm3_wmma<64><<<nBlocks, 256, 0, stream>>>(out1, t3, t4, w2, b2, out2, nRowTiles);
    }

    // ---- Heads (first 1000 nodes) ----
    head_kernel<<<blocks(1000, B), B, 0, stream>>>(out2, wch, bch, wpw, bpw,
                                                   (float*)d_out, 1000);
}